// HashGridEncoder_74783970558015
// MI455X (gfx1250) — compile-verified
//
#include <hip/hip_runtime.h>
#include <math.h>
#include <string.h>
#include <stdint.h>

// ---------------------------------------------------------------------------
// HashGrid encoder, MI455X (gfx1250, wave32).
// L2-gather-bound: one thread per (point, level).
//   lane mapping: level = tid & 15, point = tid >> 4
//   -> 16 lanes share one point's pos (single wave-coalesced b96 NT load)
//   -> one wave32 writes 2 points * 32 floats = 256B contiguous NT stores
//   -> per-level constants packed AoS {scale,res,res2,offs}: ONE b128 load
// Latents keep default RT policy (stay resident in 192MB L2); pos/out are
// streamed with non-temporal hints around them.
// ---------------------------------------------------------------------------

#define LVL      16
#define TBL      524288u      // T = 2^19
#define NPTS     262144
#define P1       2654435761u
#define P2       805459861u

typedef uint32_t u32x4 __attribute__((ext_vector_type(4)));
typedef float    f32x2 __attribute__((ext_vector_type(2)));

struct HGParams {
    u32x4   lvl[LVL];   // {bits(scale), res, res*res, offset}
    uint32_t fhl;       // first hash level
};

__global__ __launch_bounds__(256)
void hashgrid_enc_kernel(const float* __restrict__ pos,
                         const float* __restrict__ lat,
                         float* __restrict__ out,
                         HGParams p)
{
    const uint32_t tid   = blockIdx.x * 256u + threadIdx.x;
    const uint32_t point = tid >> 4;          // 16 levels per point
    const uint32_t level = tid & 15u;

    // One b128 kernarg-relative load for all per-level constants.
    const u32x4   q    = p.lvl[level];
    const float   s    = __uint_as_float(q.x);
    const uint32_t res = q.y;
    const uint32_t res2= q.z;
    const uint32_t off = q.w;
    const bool   dense = level < p.fhl;

    // pos read once per point (broadcast across 16 lanes) -> stream NT.
    const float* pp = pos + (size_t)point * 3u;
    const float px = __builtin_nontemporal_load(pp + 0);
    const float py = __builtin_nontemporal_load(pp + 1);
    const float pz = __builtin_nontemporal_load(pp + 2);

    const float sx = fmaf(px, s, 0.5f);
    const float sy = fmaf(py, s, 0.5f);
    const float sz = fmaf(pz, s, 0.5f);
    const float fx = floorf(sx);
    const float fy = floorf(sy);
    const float fz = floorf(sz);
    const float tx = sx - fx;
    const float ty = sy - fy;
    const float tz = sz - fz;

    const uint32_t x0 = (uint32_t)fx;
    const uint32_t y0 = (uint32_t)fy;
    const uint32_t z0 = (uint32_t)fz;

    // Hoisted per-dimension index contributions for both addressing modes.
    const uint32_t xd[2] = { x0,            x0 + 1u          };
    const uint32_t yd[2] = { y0 * res,      y0 * res  + res  };
    const uint32_t zd[2] = { z0 * res2,     z0 * res2 + res2 };
    const uint32_t yh[2] = { y0 * P1,       y0 * P1 + P1     };
    const uint32_t zh[2] = { z0 * P2,       z0 * P2 + P2     };

    const float wx[2] = { 1.0f - tx, tx };
    const float wy[2] = { 1.0f - ty, ty };
    const float wz[2] = { 1.0f - tz, tz };

    // Issue all 8 independent b64 gathers up front (max outstanding LOADcnt).
    f32x2 v[8];
#pragma unroll
    for (int c = 0; c < 8; ++c) {
        const uint32_t bx = (uint32_t)c & 1u;
        const uint32_t by = ((uint32_t)c >> 1) & 1u;
        const uint32_t bz = ((uint32_t)c >> 2) & 1u;
        const uint32_t di = xd[bx] + yd[by] + zd[bz];                 // dense
        const uint32_t hi = (xd[bx] ^ yh[by] ^ zh[bz]) & (TBL - 1u);  // hashed
        const uint32_t idx = (dense ? di : hi) + off;
        v[c] = *(const f32x2*)(lat + 2u * (size_t)idx);               // global_load_b64
    }

    float a0 = 0.0f, a1 = 0.0f;
#pragma unroll
    for (int c = 0; c < 8; ++c) {
        const uint32_t bx = (uint32_t)c & 1u;
        const uint32_t by = ((uint32_t)c >> 1) & 1u;
        const uint32_t bz = ((uint32_t)c >> 2) & 1u;
        const float w = wx[bx] * wy[by] * wz[bz];
        a0 = fmaf(w, v[c].x, a0);
        a1 = fmaf(w, v[c].y, a1);
    }

    // out[point][level*2 + f]; wave writes 256B contiguous, write-once -> NT.
    f32x2 r = { a0, a1 };
    __builtin_nontemporal_store(r, (f32x2*)(out + (size_t)point * (LVL * 2u) + level * 2u));
}

extern "C" void kernel_launch(void* const* d_in, const int* in_sizes, int n_in,
                              void* d_out, int out_size, void* d_ws, size_t ws_size,
                              hipStream_t stream) {
    (void)in_sizes; (void)n_in; (void)out_size; (void)d_ws; (void)ws_size;

    const float* pos = (const float*)d_in[0];
    const float* lat = (const float*)d_in[1];
    float*       out = (float*)d_out;

    // Level parameters in double precision to match the Python reference.
    HGParams p;
    const double Bg = exp((log(2048.0) - log(16.0)) / 15.0);
    uint32_t off = 0, fhl = 0;
    for (int i = 0; i < LVL; ++i) {
        const double sc = 16.0 * pow(Bg, (double)i) - 1.0;
        const float  sf = (float)sc;
        const uint32_t r = (uint32_t)ceil(sc) + 1u;
        uint32_t sbits;
        memcpy(&sbits, &sf, sizeof(sbits));
        p.lvl[i] = (u32x4){ sbits, r, r * r, off };
        uint64_t n = (uint64_t)r * r * r;
        if (n <= (uint64_t)TBL) fhl++; else n = TBL;
        off += (uint32_t)n;
    }
    p.fhl = fhl;   // = 5

    const int total = NPTS * LVL;           // one thread per (point, level)
    const int block = 256;                  // 8 wave32s
    const int grid  = total / block;        // 16384 blocks
    hashgrid_enc_kernel<<<grid, block, 0, stream>>>(pos, lat, out, p);
}